// NestedGIN_4887672783293
// MI455X (gfx1250) — compile-verified
//
#include <hip/hip_runtime.h>
#include <math.h>

// ---------------------------------------------------------------------------
// Problem constants (match reference)
// ---------------------------------------------------------------------------
#define NN   50000
#define NPAD 50048        // padded to a multiple of 64 (GEMM super-tile rows)
#define EE   800000
#define SS   5000
#define GG   64
#define HH   128
#define NCONV_EXTRA 3

typedef float v2f __attribute__((ext_vector_type(2)));
typedef float v8f __attribute__((ext_vector_type(8)));

// ---------------------------------------------------------------------------
// Utility kernels
// ---------------------------------------------------------------------------
__global__ void k_zero_i32(int* __restrict__ p, int n) {
    int i = blockIdx.x * blockDim.x + threadIdx.x;
    if (i < n) p[i] = 0;
}

__global__ void k_zero_f32(float* __restrict__ p, int n) {
    int i = blockIdx.x * blockDim.x + threadIdx.x;
    if (i < n) p[i] = 0.0f;
}

__global__ void k_copy_i32(const int* __restrict__ s, int* __restrict__ d, int n) {
    int i = blockIdx.x * blockDim.x + threadIdx.x;
    if (i < n) d[i] = s[i];
}

// ---------------------------------------------------------------------------
// CSR build: counts -> exclusive scan -> scatter edge sources by dst
// ---------------------------------------------------------------------------
__global__ void k_count_deg(const int* __restrict__ dst, int* __restrict__ counts, int e) {
    int i = blockIdx.x * blockDim.x + threadIdx.x;
    if (i < e) atomicAdd(&counts[dst[i]], 1);
}

// Single-block chunked Hillis-Steele scan over n=50000 ints (exclusive result
// in row_ptr[0..n]).  One workgroup of 1024 threads; trivial runtime.
__global__ __launch_bounds__(1024) void k_scan(const int* __restrict__ counts,
                                               int* __restrict__ row_ptr, int n) {
    __shared__ int buf[1024];
    __shared__ int carry;
    const int tid = threadIdx.x;
    if (tid == 0) { carry = 0; row_ptr[0] = 0; }
    __syncthreads();
    for (int base = 0; base < n; base += 1024) {
        int i = base + tid;
        buf[tid] = (i < n) ? counts[i] : 0;
        __syncthreads();
        for (int off = 1; off < 1024; off <<= 1) {
            int t = (tid >= off) ? buf[tid - off] : 0;
            __syncthreads();
            buf[tid] += t;
            __syncthreads();
        }
        int incl = buf[tid] + carry;
        if (i < n) row_ptr[i + 1] = incl;
        __syncthreads();
        if (tid == 0) carry += buf[1023];
        __syncthreads();
    }
}

__global__ void k_scatter_edges(const int* __restrict__ src, const int* __restrict__ dst,
                                int* __restrict__ cursor, int* __restrict__ col, int e) {
    int i = blockIdx.x * blockDim.x + threadIdx.x;
    if (i < e) {
        int pos = atomicAdd(&cursor[dst[i]], 1);
        col[pos] = src[i];
    }
}

// ---------------------------------------------------------------------------
// GIN aggregation: out[i] = h[i] + sum_{j in in-neighbors(i)} h[j]
// ---------------------------------------------------------------------------
// F=2 (first conv, on raw x): one thread per node.
__global__ void k_agg2(const float* __restrict__ x, const int* __restrict__ rowp,
                       const int* __restrict__ col, float* __restrict__ out, int n) {
    int i = blockIdx.x * blockDim.x + threadIdx.x;
    if (i >= n) return;
    float a0 = x[2 * i], a1 = x[2 * i + 1];
    int e0 = rowp[i], e1 = rowp[i + 1];
    for (int e = e0; e < e1; ++e) {
        int s = col[e];
        a0 += x[2 * s];
        a1 += x[2 * s + 1];
    }
    out[2 * i]     = a0;
    out[2 * i + 1] = a1;
}

// F=128: one wave (32 lanes) per node, float4 per lane (512B coalesced rows).
// Edges processed 8 at a time so 8 index loads + 8 row gathers are in flight
// per clause (MLP vs L2 latency); next index chunk is prefetched.
__global__ __launch_bounds__(256) void k_agg128(const float* __restrict__ h,
                                                const int* __restrict__ rowp,
                                                const int* __restrict__ col,
                                                float* __restrict__ out, int n) {
    int node = blockIdx.x * 8 + (threadIdx.x >> 5);
    if (node >= n) return;
    int lane = threadIdx.x & 31;
    const float4* hv = (const float4*)h;
    float4 acc = hv[(size_t)node * 32 + lane];
    int e0 = rowp[node], e1 = rowp[node + 1];
    int e = e0;
    for (; e + 8 <= e1; e += 8) {
        __builtin_prefetch(&col[e + 8], 0, 0);      // global_prefetch_b8
        int i0 = col[e + 0], i1 = col[e + 1], i2 = col[e + 2], i3 = col[e + 3];
        int i4 = col[e + 4], i5 = col[e + 5], i6 = col[e + 6], i7 = col[e + 7];
        float4 v0 = hv[(size_t)i0 * 32 + lane];
        float4 v1 = hv[(size_t)i1 * 32 + lane];
        float4 v2 = hv[(size_t)i2 * 32 + lane];
        float4 v3 = hv[(size_t)i3 * 32 + lane];
        float4 v4 = hv[(size_t)i4 * 32 + lane];
        float4 v5 = hv[(size_t)i5 * 32 + lane];
        float4 v6 = hv[(size_t)i6 * 32 + lane];
        float4 v7 = hv[(size_t)i7 * 32 + lane];
        acc.x += v0.x + v1.x + v2.x + v3.x + v4.x + v5.x + v6.x + v7.x;
        acc.y += v0.y + v1.y + v2.y + v3.y + v4.y + v5.y + v6.y + v7.y;
        acc.z += v0.z + v1.z + v2.z + v3.z + v4.z + v5.z + v6.z + v7.z;
        acc.w += v0.w + v1.w + v2.w + v3.w + v4.w + v5.w + v6.w + v7.w;
    }
    for (; e < e1; ++e) {
        int s = col[e];
        float4 v = hv[(size_t)s * 32 + lane];
        acc.x += v.x; acc.y += v.y; acc.z += v.z; acc.w += v.w;
    }
    ((float4*)out)[(size_t)node * 32 + lane] = acc;
}

// ---------------------------------------------------------------------------
// (N x 2) @ (2 x 128) + bias, ReLU — trivial VALU kernel, thread = (node, 4 cols)
// ---------------------------------------------------------------------------
__global__ void k_gemm2(const float* __restrict__ xa, const float* __restrict__ W,
                        const float* __restrict__ b, float* __restrict__ out, int n) {
    int t = blockIdx.x * blockDim.x + threadIdx.x;
    if (t >= n * 32) return;
    int node = t >> 5;
    int c4   = t & 31;
    float a0 = xa[2 * node], a1 = xa[2 * node + 1];
    const float4* W4 = (const float4*)W;
    const float4* b4 = (const float4*)b;
    float4 w0 = W4[c4], w1 = W4[32 + c4], bb = b4[c4];
    float4 r;
    r.x = fmaxf(a0 * w0.x + a1 * w1.x + bb.x, 0.0f);
    r.y = fmaxf(a0 * w0.y + a1 * w1.y + bb.y, 0.0f);
    r.z = fmaxf(a0 * w0.z + a1 * w1.z + bb.z, 0.0f);
    r.w = fmaxf(a0 * w0.w + a1 * w1.w + bb.w, 0.0f);
    ((float4*)out)[t] = r;
}

// ---------------------------------------------------------------------------
// (M x 128) @ (128 x 128) + bias (+ReLU) via V_WMMA_F32_16X16X4_F32.
// grid.x = M/64 ; block = 256 (8 waves).  Block owns a 64x128 output
// super-tile; wave w owns N-tile w and 4 M-tiles -> 4 independent WMMA
// accumulator chains (hides the f32 WMMA D->C latency) and 4x B-frag reuse.
// A panel (64x128 f32 = 32KB) staged in LDS; W (64KB) stays L2/L0-resident.
// A frag (16x4 f32): lane<16 -> {K0,K1}, lane>=16 -> {K2,K3}   (ISA 7.12.2)
// C tile: VGPR i -> row i (+8 for lane>=16), col = lane&15.
// EXEC is all-ones (no divergence) as WMMA requires.
// ---------------------------------------------------------------------------
__global__ __launch_bounds__(256) void k_gemm128_wmma(const float* __restrict__ A,
                                                      const float* __restrict__ W,
                                                      const float* __restrict__ bias,
                                                      float* __restrict__ out,
                                                      int relu) {
    __shared__ float As[64 * 128];              // 32 KB
    const int    tid     = threadIdx.x;
    const size_t rowbase = (size_t)blockIdx.x * 64;

    const float4* Ag  = (const float4*)(A + rowbase * 128);
    float4*       As4 = (float4*)As;
#pragma unroll
    for (int i = 0; i < 8; ++i) As4[tid + 256 * i] = Ag[tid + 256 * i];
    __syncthreads();

    const int wave = tid >> 5;
    const int lane = tid & 31;
    const int r    = lane & 15;
    const int kh   = (lane >> 4) << 1;   // 0 for lanes 0-15, 2 for lanes 16-31
    const int ncol = wave * 16 + r;

    v8f c0 = {}, c1 = {}, c2 = {}, c3 = {};
#pragma unroll
    for (int k = 0; k < 128; k += 4) {
        v2f b;
        b[0] = W[(size_t)(k + kh)     * 128 + ncol];
        b[1] = W[(size_t)(k + kh + 1) * 128 + ncol];
        v2f a0, a1, a2, a3;
        float2 t;
        t = *(const float2*)&As[(r     ) * 128 + k + kh]; a0[0] = t.x; a0[1] = t.y;
        t = *(const float2*)&As[(r + 16) * 128 + k + kh]; a1[0] = t.x; a1[1] = t.y;
        t = *(const float2*)&As[(r + 32) * 128 + k + kh]; a2[0] = t.x; a2[1] = t.y;
        t = *(const float2*)&As[(r + 48) * 128 + k + kh]; a3[0] = t.x; a3[1] = t.y;
        c0 = __builtin_amdgcn_wmma_f32_16x16x4_f32(false, a0, false, b, (short)0, c0, false, false);
        c1 = __builtin_amdgcn_wmma_f32_16x16x4_f32(false, a1, false, b, (short)0, c1, false, false);
        c2 = __builtin_amdgcn_wmma_f32_16x16x4_f32(false, a2, false, b, (short)0, c2, false, false);
        c3 = __builtin_amdgcn_wmma_f32_16x16x4_f32(false, a3, false, b, (short)0, c3, false, false);
    }

    const float bc = bias[ncol];
    const int   rb = (lane >> 4) * 8;
    float*      o  = out + rowbase * 128 + ncol;
#pragma unroll
    for (int i = 0; i < 8; ++i) {
        float v;
        v = c0[i] + bc; if (relu) v = fmaxf(v, 0.0f); o[(size_t)(rb + i     ) * 128] = v;
        v = c1[i] + bc; if (relu) v = fmaxf(v, 0.0f); o[(size_t)(rb + i + 16) * 128] = v;
        v = c2[i] + bc; if (relu) v = fmaxf(v, 0.0f); o[(size_t)(rb + i + 32) * 128] = v;
        v = c3[i] + bc; if (relu) v = fmaxf(v, 0.0f); o[(size_t)(rb + i + 48) * 128] = v;
    }
}

// ---------------------------------------------------------------------------
// Segment-sum pooling: thread = (row, feature), atomic f32 add into segment.
// ---------------------------------------------------------------------------
__global__ void k_pool(const float* __restrict__ in, const int* __restrict__ seg,
                       float* __restrict__ out, int total) {
    int t = blockIdx.x * blockDim.x + threadIdx.x;
    if (t >= total) return;
    int row = t >> 7;
    int f   = t & 127;
    atomicAdd(&out[(size_t)seg[row] * 128 + f], in[t]);
}

// ---------------------------------------------------------------------------
// log_softmax over 64 rows of 128: one block per row.
// ---------------------------------------------------------------------------
__global__ __launch_bounds__(128) void k_logsoftmax(const float* __restrict__ in,
                                                    float* __restrict__ out) {
    __shared__ float red[128];
    const int row = blockIdx.x;
    const int t   = threadIdx.x;
    float v = in[(size_t)row * 128 + t];
    red[t] = v;
    __syncthreads();
    for (int off = 64; off > 0; off >>= 1) {
        if (t < off) red[t] = fmaxf(red[t], red[t + off]);
        __syncthreads();
    }
    float m = red[0];
    __syncthreads();
    red[t] = __expf(v - m);
    __syncthreads();
    for (int off = 64; off > 0; off >>= 1) {
        if (t < off) red[t] += red[t + off];
        __syncthreads();
    }
    float lse = logf(red[0]);
    out[(size_t)row * 128 + t] = v - m - lse;
}

// ---------------------------------------------------------------------------
// Host launch
// ---------------------------------------------------------------------------
extern "C" void kernel_launch(void* const* d_in, const int* in_sizes, int n_in,
                              void* d_out, int out_size, void* d_ws, size_t ws_size,
                              hipStream_t stream) {
    (void)in_sizes; (void)n_in; (void)out_size; (void)ws_size;

    const float* x    = (const float*)d_in[0];
    const int*   ei   = (const int*)d_in[1];
    const int*   src  = ei;
    const int*   dst  = ei + EE;
    const int*   n2s  = (const int*)d_in[2];
    const int*   s2g  = (const int*)d_in[3];
    const float* W1a  = (const float*)d_in[4];
    const float* b1a  = (const float*)d_in[5];
    const float* W1b  = (const float*)d_in[6];
    const float* b1b  = (const float*)d_in[7];
    const float* cWa  = (const float*)d_in[8];
    const float* cba  = (const float*)d_in[9];
    const float* cWb  = (const float*)d_in[10];
    const float* cbb  = (const float*)d_in[11];
    const float* l1W  = (const float*)d_in[12];
    const float* l1b  = (const float*)d_in[13];
    const float* l2W  = (const float*)d_in[14];
    const float* l2b  = (const float*)d_in[15];
    float*       outp = (float*)d_out;

    // Workspace carve-up (256B aligned slices).
    char*  base = (char*)d_ws;
    size_t off  = 0;
    auto carve = [&](size_t bytes) -> char* {
        char* p = base + off;
        off = (off + bytes + 255) & ~(size_t)255;
        return p;
    };
    float* hA      = (float*)carve((size_t)NPAD * HH * 4);   // ~25.6 MB each
    float* hB      = (float*)carve((size_t)NPAD * HH * 4);
    float* hC      = (float*)carve((size_t)NPAD * HH * 4);
    float* xagg    = (float*)carve((size_t)NN * 2 * 4);
    float* subpool = (float*)carve((size_t)SS * HH * 4);
    float* gpool   = (float*)carve((size_t)GG * HH * 4);
    float* glin1   = (float*)carve((size_t)GG * HH * 4);
    float* glin2   = (float*)carve((size_t)GG * HH * 4);
    int*   counts  = (int*)carve((size_t)NN * 4);
    int*   row_ptr = (int*)carve((size_t)(NN + 1) * 4);
    int*   cursor  = (int*)carve((size_t)NN * 4);
    int*   col     = (int*)carve((size_t)EE * 4);

    const int T = 256;
    auto blk = [](int n, int t) { return (n + t - 1) / t; };
    const int PADELEMS = (NPAD - NN) * HH;   // 48*128, keep pad rows finite

    // ---- keep GEMM pad rows (50000..50047) finite; never consumed ----
    k_zero_f32<<<blk(PADELEMS, T), T, 0, stream>>>(hA + (size_t)NN * HH, PADELEMS);
    k_zero_f32<<<blk(PADELEMS, T), T, 0, stream>>>(hB + (size_t)NN * HH, PADELEMS);
    k_zero_f32<<<blk(PADELEMS, T), T, 0, stream>>>(hC + (size_t)NN * HH, PADELEMS);

    // ---- CSR build (per-launch, deterministic work) ----
    k_zero_i32<<<blk(NN, T), T, 0, stream>>>(counts, NN);
    k_count_deg<<<blk(EE, T), T, 0, stream>>>(dst, counts, EE);
    k_scan<<<1, 1024, 0, stream>>>(counts, row_ptr, NN);
    k_copy_i32<<<blk(NN, T), T, 0, stream>>>(row_ptr, cursor, NN);
    k_scatter_edges<<<blk(EE, T), T, 0, stream>>>(src, dst, cursor, col, EE);

    // ---- conv 1: aggregate x (F=2), 2->128 linear, 128->128 WMMA linear ----
    k_agg2<<<blk(NN, T), T, 0, stream>>>(x, row_ptr, col, xagg, NN);
    k_gemm2<<<blk(NN * 32, T), T, 0, stream>>>(xagg, W1a, b1a, hA, NN);
    k_gemm128_wmma<<<NPAD / 64, T, 0, stream>>>(hA, W1b, b1b, hB, 1);

    // ---- convs 2..4: CSR gather-aggregate + two WMMA GEMMs each ----
    for (int i = 0; i < NCONV_EXTRA; ++i) {
        const float* Wa = cWa + (size_t)i * HH * HH;
        const float* ba = cba + (size_t)i * HH;
        const float* Wb = cWb + (size_t)i * HH * HH;
        const float* bb = cbb + (size_t)i * HH;
        k_agg128<<<blk(NN, 8), T, 0, stream>>>(hB, row_ptr, col, hA, NN);
        k_gemm128_wmma<<<NPAD / 64, T, 0, stream>>>(hA, Wa, ba, hC, 1);
        k_gemm128_wmma<<<NPAD / 64, T, 0, stream>>>(hC, Wb, bb, hB, 1);
    }

    // ---- nested pooling: nodes -> subgraphs -> graphs ----
    k_zero_f32<<<blk(SS * HH, T), T, 0, stream>>>(subpool, SS * HH);
    k_pool<<<blk(NN * HH, T), T, 0, stream>>>(hB, n2s, subpool, NN * HH);
    k_zero_f32<<<blk(GG * HH, T), T, 0, stream>>>(gpool, GG * HH);
    k_pool<<<blk(SS * HH, T), T, 0, stream>>>(subpool, s2g, gpool, SS * HH);

    // ---- heads: lin1+ReLU, lin2, log_softmax ----
    k_gemm128_wmma<<<GG / 64, T, 0, stream>>>(gpool, l1W, l1b, glin1, 1);
    k_gemm128_wmma<<<GG / 64, T, 0, stream>>>(glin1, l2W, l2b, glin2, 0);
    k_logsoftmax<<<GG, 128, 0, stream>>>(glin2, outp);
}